// AttentionResidual_40286793236988
// MI455X (gfx1250) — compile-verified
//
#include <hip/hip_runtime.h>
#include <math.h>

#define N_SLICES 12
#define D_DIM    2048
#define TPB      256
#define EPS_K    1e-6f

typedef unsigned int v4u __attribute__((ext_vector_type(4)));
typedef int          v8i __attribute__((ext_vector_type(8)));
typedef int          v4i __attribute__((ext_vector_type(4)));

#if __has_builtin(__builtin_amdgcn_tensor_load_to_lds)
#define USE_TDM 1
#else
#define USE_TDM 0
#endif

// One workgroup per (b,s) position. The Tensor Data Mover DMAs the 12x2048
// fp32 tile (row stride = BS*D elements) straight into LDS in ONE instruction
// (TENSORcnt), then we compute RMS-dot logits, softmax over depth, and the
// weighted sum. V crosses HBM exactly once: ~872 MB total => ~37 us at 23.3 TB/s.
__global__ __launch_bounds__(TPB) void depth_attn_kernel(
    const float* __restrict__ V,
    const float* __restrict__ norm_scale,
    const float* __restrict__ query,
    float* __restrict__ out,
    int BS)
{
    extern __shared__ __align__(16) float smem[];
    float* lds_v   = smem;                          // N_SLICES * D_DIM floats (96 KB)
    float* red_ss  = smem + N_SLICES * D_DIM;       // N_SLICES * 8
    float* red_dt  = red_ss + N_SLICES * 8;         // N_SLICES * 8
    float* s_logit = red_dt + N_SLICES * 8;         // N_SLICES

    const int tid  = threadIdx.x;
    const int lane = tid & 31;
    const int wid  = tid >> 5;

    const long long sidx         = blockIdx.x;                 // 0..BS-1
    const long long slice_stride = (long long)BS * D_DIM;      // elements between depth slices
    const float*    vbase        = V + sidx * (long long)D_DIM;
    const unsigned  lds_base     = (unsigned)(size_t)lds_v;    // LDS offset = low 32 bits of the generic pointer

#if USE_TDM
    // ---- one TDM descriptor loads the whole 12 x 8KB tile into LDS ----
    if (wid == 0) {
        const unsigned long long ga = (unsigned long long)(const void*)vbase;
        v4u g0;
        g0[0] = 1u;                                          // [1:0] count=1 (valid user D#)
        g0[1] = lds_base;                                    // [63:32] lds_addr (bytes)
        g0[2] = (unsigned)(ga & 0xFFFFFFFFu);                // [95:64] global_addr lo
        g0[3] = (unsigned)((ga >> 32) & 0x01FFFFFFu)         // [120:96] global_addr hi (57-bit)
              | (2u << 30);                                  // [127:126] type=2 ("image")
        v8i g1;
        g1[0] = (int)(2u << 16);                             // [17:16] data_size=2 -> 4B elems
        g1[1] = (int)((unsigned)D_DIM << 16);                // [63:48] tensor_dim0 = 2048
        g1[2] = (int)((unsigned)N_SLICES << 16);             // [95:80] tensor_dim1 = 12
        g1[3] = (int)((unsigned)D_DIM << 16);                // [127:112] tile_dim0 = 2048
        g1[4] = (int)N_SLICES;                               // [143:128] tile_dim1 = 12 (tile_dim2=0)
        g1[5] = (int)slice_stride;                           // [191:160] tensor_dim0_stride lo (=BS*D)
        g1[6] = 0;                                           // stride hi / tensor_dim1_stride lo
        g1[7] = 0;                                           // tensor_dim1_stride hi
        const v4i gz4 = {0, 0, 0, 0};                        // groups 2/3 unused (2D tile)
        const v8i gz8 = {0, 0, 0, 0, 0, 0, 0, 0};            // trailing group (unused, zero)
        __builtin_amdgcn_tensor_load_to_lds(g0, g1, gz4, gz4, gz8, 0);
    }
#else
    // ---- fallback: per-wave async global->LDS b128 copies (ASYNCcnt) ----
    {
        const char* g = (const char*)vbase + tid * 16;
        unsigned    l = lds_base + (unsigned)(tid * 16);
        const long long sbytes = slice_stride * 4;
        #pragma unroll
        for (int n = 0; n < N_SLICES; ++n) {
            asm volatile("global_load_async_to_lds_b128 %0, %1, off"
                         :: "v"(l), "v"(g) : "memory");
            asm volatile("global_load_async_to_lds_b128 %0, %1, off offset:4096"
                         :: "v"(l), "v"(g) : "memory");
            g += sbytes;
            l += (unsigned)(D_DIM * 4);
        }
    }
#endif

    // ---- overlap with DMA: build projection vector w = norm_scale * query ----
    const int d0 = tid * 4;
    const int d1 = 1024 + tid * 4;
    float4 ns0 = *(const float4*)(norm_scale + d0);
    float4 ns1 = *(const float4*)(norm_scale + d1);
    float4 q0  = *(const float4*)(query + d0);
    float4 q1  = *(const float4*)(query + d1);
    float4 w0  = make_float4(ns0.x * q0.x, ns0.y * q0.y, ns0.z * q0.z, ns0.w * q0.w);
    float4 w1  = make_float4(ns1.x * q1.x, ns1.y * q1.y, ns1.z * q1.z, ns1.w * q1.w);

#if USE_TDM
    if (wid == 0) __builtin_amdgcn_s_wait_tensorcnt(0);      // wave 0's DMA landed in LDS
#else
    asm volatile("s_wait_asynccnt 0" ::: "memory");          // my wave's copies landed
#endif
    __syncthreads();                                         // publish LDS fill to all waves

    // ---- phase 1: per-slice sum(v*v) and dot(v, w), block-reduced ----
    #pragma unroll
    for (int n = 0; n < N_SLICES; ++n) {
        const float4 a = *(const float4*)&lds_v[n * D_DIM + d0];
        const float4 b = *(const float4*)&lds_v[n * D_DIM + d1];
        float ss = a.x * a.x;
        ss = fmaf(a.y, a.y, ss); ss = fmaf(a.z, a.z, ss); ss = fmaf(a.w, a.w, ss);
        ss = fmaf(b.x, b.x, ss); ss = fmaf(b.y, b.y, ss);
        ss = fmaf(b.z, b.z, ss); ss = fmaf(b.w, b.w, ss);
        float dt = a.x * w0.x;
        dt = fmaf(a.y, w0.y, dt); dt = fmaf(a.z, w0.z, dt); dt = fmaf(a.w, w0.w, dt);
        dt = fmaf(b.x, w1.x, dt); dt = fmaf(b.y, w1.y, dt);
        dt = fmaf(b.z, w1.z, dt); dt = fmaf(b.w, w1.w, dt);
        #pragma unroll
        for (int m = 16; m >= 1; m >>= 1) {
            ss += __shfl_xor(ss, m, 32);
            dt += __shfl_xor(dt, m, 32);
        }
        if (lane == 0) { red_ss[n * 8 + wid] = ss; red_dt[n * 8 + wid] = dt; }
    }
    __syncthreads();

    if (tid < N_SLICES) {
        float ss = 0.f, dt = 0.f;
        #pragma unroll
        for (int wv = 0; wv < 8; ++wv) { ss += red_ss[tid * 8 + wv]; dt += red_dt[tid * 8 + wv]; }
        const float rms = sqrtf(ss * (1.0f / (float)D_DIM) + EPS_K);
        s_logit[tid] = dt / rms;
    }
    __syncthreads();

    // ---- softmax over depth (12 values, computed redundantly per thread) ----
    float wgt[N_SLICES];
    float mx = -INFINITY;
    #pragma unroll
    for (int n = 0; n < N_SLICES; ++n) { wgt[n] = s_logit[n]; mx = fmaxf(mx, wgt[n]); }
    float sum = 0.f;
    #pragma unroll
    for (int n = 0; n < N_SLICES; ++n) { wgt[n] = expf(wgt[n] - mx); sum += wgt[n]; }
    const float inv = 1.0f / sum;

    // ---- phase 2: weighted sum over depth from LDS ----
    float4 acc0 = make_float4(0.f, 0.f, 0.f, 0.f);
    float4 acc1 = make_float4(0.f, 0.f, 0.f, 0.f);
    #pragma unroll
    for (int n = 0; n < N_SLICES; ++n) {
        const float  wn = wgt[n] * inv;
        const float4 a  = *(const float4*)&lds_v[n * D_DIM + d0];
        const float4 b  = *(const float4*)&lds_v[n * D_DIM + d1];
        acc0.x = fmaf(wn, a.x, acc0.x); acc0.y = fmaf(wn, a.y, acc0.y);
        acc0.z = fmaf(wn, a.z, acc0.z); acc0.w = fmaf(wn, a.w, acc0.w);
        acc1.x = fmaf(wn, b.x, acc1.x); acc1.y = fmaf(wn, b.y, acc1.y);
        acc1.z = fmaf(wn, b.z, acc1.z); acc1.w = fmaf(wn, b.w, acc1.w);
    }

    float* o = out + sidx * (long long)D_DIM;
    *(float4*)(o + d0) = acc0;
    *(float4*)(o + d1) = acc1;
}

extern "C" void kernel_launch(void* const* d_in, const int* in_sizes, int n_in,
                              void* d_out, int out_size, void* d_ws, size_t ws_size,
                              hipStream_t stream) {
    const float* V  = (const float*)d_in[0];   // [N,B,S,D] fp32
    const float* ns = (const float*)d_in[1];   // [D]
    const float* q  = (const float*)d_in[2];   // [D]
    float*       o  = (float*)d_out;           // [B,S,D]

    const int D  = in_sizes[1];                // 2048 (matches D_DIM)
    const int BS = out_size / D;               // 8192 positions

    const size_t shmem = (size_t)(N_SLICES * D_DIM) * sizeof(float)   // slice cache
                       + (size_t)(N_SLICES * 8 * 2) * sizeof(float)   // reductions
                       + (size_t)N_SLICES * sizeof(float);            // logits

    depth_attn_kernel<<<BS, TPB, shmem, stream>>>(V, ns, q, o, BS);
}